// GlobalBranch_88656714924656
// MI455X (gfx1250) — compile-verified
//
#include <hip/hip_runtime.h>
#include <math.h>

#define BB 8
#define DD 64
#define D2 128
#define HH 128
#define WW 128
#define HWN (HH*WW)
#define GG 4
#define GK (D2/GG)
#define EPS 1e-5f

typedef __attribute__((ext_vector_type(2))) float v2f;
typedef __attribute__((ext_vector_type(8))) float v8f;

// D = A(16x4 f32) * B(4x16 f32) + C(16x16 f32), full fp32 precision.
__device__ __forceinline__ v8f wmma4(v2f a, v2f b, v8f c) {
  return __builtin_amdgcn_wmma_f32_16x16x4_f32(false, a, false, b, (short)0, c,
                                               false, false);
}
__device__ __forceinline__ v2f ld2(const float* p) { return *(const v2f*)p; }

// K-pair interleaved LDS layout for WMMA B-sources: (k,n) and (k+1,n) adjacent.
// Pair-row stride 288 (== 32 mod 64 banks) -> conflict-free full-wave b64 loads.
#define XI(k, n) ((((k) >> 1) * 288) + (((n)) << 1) + ((k) & 1))

__device__ __forceinline__ float gelu_exact(float x) {
  return 0.5f * x * (1.f + erff(x * 0.70710678118654752f));
}
__device__ __forceinline__ float silu(float x) { return x / (1.f + expf(-x)); }

// ---------------------------------------------------------------------------
// Init: DCT-II orthonormal matrix Dm[k][n], 128x128
__global__ void k_init_dct(float* __restrict__ Dm) {
  int t = blockIdx.x * 256 + threadIdx.x;
  if (t >= 128 * 128) return;
  int k = t >> 7, n = t & 127;
  float s = (k == 0) ? sqrtf(1.f / 128.f) : sqrtf(2.f / 128.f);
  Dm[t] = cosf(3.14159265358979323846f * (2.f * n + 1.f) * k / 256.f) * s;
}

// Init: spatial filter (needs global max over the 128x128 grid)
__global__ void k_init_filter(const float* __restrict__ beta_p,
                              float* __restrict__ filt) {
  __shared__ float red[256];
  const float beta = beta_p[0];
  const float dn = rsqrtf((float)((HH - 1) * (HH - 1) + (WW - 1) * (WW - 1)));
  float mymax = 0.f;
  for (int i = threadIdx.x; i < HWN; i += 256) {
    int y = i >> 7, x = i & 127;
    float nd = sqrtf((float)(x * x + y * y)) * dn;
    float xv = 3.f * (nd - 0.2f);
    float sw = xv / (1.f + expf(-beta * xv));
    mymax = fmaxf(mymax, fabsf(sw));
  }
  red[threadIdx.x] = mymax;
  __syncthreads();
  for (int s = 128; s > 0; s >>= 1) {
    if (threadIdx.x < s) red[threadIdx.x] = fmaxf(red[threadIdx.x], red[threadIdx.x + s]);
    __syncthreads();
  }
  const float me = red[0] + 1e-8f;
  for (int i = threadIdx.x; i < HWN; i += 256) {
    int y = i >> 7, x = i & 127;
    float nd = sqrtf((float)(x * x + y * y)) * dn;
    float xv = 3.f * (nd - 0.2f);
    float sw = xv / (1.f + expf(-beta * xv));
    float f = 1.f + 0.8f * sw / me;
    filt[i] = fminf(fmaxf(f, 0.7f), 2.2f);
  }
}

// ---------------------------------------------------------------------------
// conv1x1 64->128 + exact GELU, via fp32 WMMA. grid = B * (HW/128), 256 thr.
__global__ void k_ci(const float* __restrict__ x, const float* __restrict__ w,
                     const float* __restrict__ bias, float* __restrict__ x1) {
  extern __shared__ float sm[];
  float* Wl = sm;            // [128][68] row-major (A source)
  float* Xl = sm + 128 * 68; // 32 pairs x 288, interleaved (B source)
  const int tid = threadIdx.x;
  const int batch = blockIdx.x >> 7;
  const int p0 = (blockIdx.x & 127) * 128;
  for (int i = tid; i < 128 * 64; i += 256) Wl[(i >> 6) * 68 + (i & 63)] = w[i];
  for (int i = tid; i < 64 * 128; i += 256) {
    int c = i >> 7, px = i & 127;
    Xl[XI(c, px)] = x[((size_t)batch * DD + c) * HWN + p0 + px];
  }
  __syncthreads();
  const int wave = tid >> 5, lane = tid & 31;
  const int lh = lane & 15, kh = (lane >> 4) * 2, hi = lane >> 4;
  const int m0 = wave * 16;
  v8f zero = {};
  v8f acc[8];
#pragma unroll
  for (int n = 0; n < 8; ++n) acc[n] = zero;
  // software-pipelined: prefetch K-step k0+4 before the WMMA chain of k0.
  v2f a_c = ld2(&Wl[(m0 + lh) * 68 + kh]);
  v2f bf_c[8];
#pragma unroll
  for (int n = 0; n < 8; ++n) bf_c[n] = ld2(&Xl[XI(kh, n * 16 + lh)]);
  for (int k0 = 0; k0 < 60; k0 += 4) {
    const int kn = k0 + 4;
    v2f a_n = ld2(&Wl[(m0 + lh) * 68 + kn + kh]);
    v2f bf_n[8];
#pragma unroll
    for (int n = 0; n < 8; ++n) bf_n[n] = ld2(&Xl[XI(kn + kh, n * 16 + lh)]);
#pragma unroll
    for (int n = 0; n < 8; ++n) acc[n] = wmma4(a_c, bf_c[n], acc[n]);
    a_c = a_n;
#pragma unroll
    for (int n = 0; n < 8; ++n) bf_c[n] = bf_n[n];
  }
#pragma unroll
  for (int n = 0; n < 8; ++n) acc[n] = wmma4(a_c, bf_c[n], acc[n]);
#pragma unroll
  for (int n = 0; n < 8; ++n)
#pragma unroll
    for (int r = 0; r < 8; ++r) {
      int o = m0 + r + hi * 8, px = n * 16 + lh;
      float v = acc[n][r] + bias[o];
      x1[((size_t)batch * D2 + o) * HWN + p0 + px] = gelu_exact(v);
    }
}

// ---------------------------------------------------------------------------
// Separable DCT per (b,c) image, fully in LDS: T1 = A*X then OUT = T1*B.
// forward (inverse=0): A=Dm, B=Dm^T, fused spatial filter on store.
//   Dl row-major [128][132]: A contiguous in k; B^T reads (n,k),(n,k+1) contiguous.
// inverse (inverse=1): A=Dm^T, B=Dm.
//   Dl interleaved: (k,m),(k+1,m) and (k,n),(k+1,n) contiguous.
// X input interleaved; T1 rewritten row-major [128][132] (A source for stage 2).
// grid = B*128 images, 256 thr.
__global__ void k_dct(const float* __restrict__ in, float* __restrict__ out,
                      const float* __restrict__ Dm, const float* __restrict__ filt,
                      int inverse) {
  extern __shared__ float sm[];
  float* Dl = sm;           // 18432 floats
  float* Xl = sm + 18432;   // 18432 floats
  const int tid = threadIdx.x;
  const int img = blockIdx.x;
  for (int i = tid; i < 128 * 128; i += 256) {
    int r = i >> 7, c = i & 127;
    if (!inverse) Dl[r * 132 + c] = Dm[i];
    else          Dl[XI(r, c)] = Dm[i];
    Xl[XI(r, c)] = in[(size_t)img * HWN + i];
  }
  __syncthreads();
  const int wave = tid >> 5, lane = tid & 31;
  const int lh = lane & 15, kh = (lane >> 4) * 2, hi = lane >> 4;
  const int m0 = wave * 16;
  v8f zero = {};
  v8f acc[8];
#pragma unroll
  for (int n = 0; n < 8; ++n) acc[n] = zero;
  // ---- stage 1: T1 = (inverse? Dm^T : Dm) * X, software-pipelined ----
  {
    v2f a_c, bf_c[8];
    if (!inverse) a_c = ld2(&Dl[(m0 + lh) * 132 + kh]);
    else          a_c = ld2(&Dl[XI(kh, m0 + lh)]);
#pragma unroll
    for (int n = 0; n < 8; ++n) bf_c[n] = ld2(&Xl[XI(kh, n * 16 + lh)]);
    for (int k0 = 0; k0 < 124; k0 += 4) {
      const int kn = k0 + 4;
      v2f a_n, bf_n[8];
      if (!inverse) a_n = ld2(&Dl[(m0 + lh) * 132 + kn + kh]);
      else          a_n = ld2(&Dl[XI(kn + kh, m0 + lh)]);
#pragma unroll
      for (int n = 0; n < 8; ++n) bf_n[n] = ld2(&Xl[XI(kn + kh, n * 16 + lh)]);
#pragma unroll
      for (int n = 0; n < 8; ++n) acc[n] = wmma4(a_c, bf_c[n], acc[n]);
      a_c = a_n;
#pragma unroll
      for (int n = 0; n < 8; ++n) bf_c[n] = bf_n[n];
    }
#pragma unroll
    for (int n = 0; n < 8; ++n) acc[n] = wmma4(a_c, bf_c[n], acc[n]);
  }
  __syncthreads(); // everyone done reading X
#pragma unroll
  for (int n = 0; n < 8; ++n)
#pragma unroll
    for (int r = 0; r < 8; ++r)
      Xl[(m0 + r + hi * 8) * 132 + n * 16 + lh] = acc[n][r];
  __syncthreads();
  // ---- stage 2: OUT = T1 * (inverse? Dm : Dm^T), software-pipelined ----
#pragma unroll
  for (int n = 0; n < 8; ++n) acc[n] = zero;
  {
    v2f a_c = ld2(&Xl[(m0 + lh) * 132 + kh]);
    v2f bf_c[8];
    if (!inverse) {
#pragma unroll
      for (int n = 0; n < 8; ++n) bf_c[n] = ld2(&Dl[(n * 16 + lh) * 132 + kh]);
    } else {
#pragma unroll
      for (int n = 0; n < 8; ++n) bf_c[n] = ld2(&Dl[XI(kh, n * 16 + lh)]);
    }
    for (int k0 = 0; k0 < 124; k0 += 4) {
      const int kn = k0 + 4;
      v2f a_n = ld2(&Xl[(m0 + lh) * 132 + kn + kh]);
      v2f bf_n[8];
      if (!inverse) {
#pragma unroll
        for (int n = 0; n < 8; ++n) bf_n[n] = ld2(&Dl[(n * 16 + lh) * 132 + kn + kh]);
      } else {
#pragma unroll
        for (int n = 0; n < 8; ++n) bf_n[n] = ld2(&Dl[XI(kn + kh, n * 16 + lh)]);
      }
#pragma unroll
      for (int n = 0; n < 8; ++n) acc[n] = wmma4(a_c, bf_c[n], acc[n]);
      a_c = a_n;
#pragma unroll
      for (int n = 0; n < 8; ++n) bf_c[n] = bf_n[n];
    }
#pragma unroll
    for (int n = 0; n < 8; ++n) acc[n] = wmma4(a_c, bf_c[n], acc[n]);
  }
#pragma unroll
  for (int n = 0; n < 8; ++n)
#pragma unroll
    for (int r = 0; r < 8; ++r) {
      int row = m0 + r + hi * 8, col = n * 16 + lh;
      float v = acc[n][r];
      if (!inverse) v *= filt[row * WW + col];
      out[(size_t)img * HWN + row * WW + col] = v;
    }
}

// ---------------------------------------------------------------------------
// depthwise 3x3 SAME conv + bias + identity. grid = B*D2*HW/256.
__global__ void k_fpe(const float* __restrict__ dct, const float* __restrict__ wf,
                      const float* __restrict__ bf, float* __restrict__ dyn) {
  size_t idx = (size_t)blockIdx.x * 256 + threadIdx.x;
  int hw = (int)(idx & (HWN - 1));
  int bc = (int)(idx >> 14);
  int c = bc & 127;
  int h = hw >> 7, w = hw & 127;
  const float* img = dct + ((size_t)bc << 14);
  float s = 0.f;
#pragma unroll
  for (int dy = -1; dy <= 1; ++dy)
#pragma unroll
    for (int dx = -1; dx <= 1; ++dx) {
      int hh = h + dy, ww = w + dx;
      if (hh >= 0 && hh < HH && ww >= 0 && ww < WW)
        s += img[hh * WW + ww] * wf[c * 9 + (dy + 1) * 3 + (dx + 1)];
    }
  dyn[idx] = s + bf[c] + img[hw];
}

// ---------------------------------------------------------------------------
// dyw softmax + grouped fdc conv (WMMA) + fuse add; per-group accumulators,
// dyw scale applied after accumulation (linearity). Fuse written IN PLACE over
// dct_feat. grid = B*(HW/128), 256 thr.
__global__ void k_mix(const float* __restrict__ dyn, float* __restrict__ dct_fuse,
                      const float* __restrict__ wtw, const float* __restrict__ wtb,
                      const float* __restrict__ fdcw, const float* __restrict__ fdcb) {
  extern __shared__ float sm[];
  float* Xl = sm;               // interleaved 64 pairs x 288 (dct_dyn tile)
  float* Fl = Xl + 18432;       // [512][36] fdc weights (g*128+o, k)
  float* dyw = Fl + 512 * 36;   // [4][128]
  float* wtl = dyw + 512;       // [4][128]
  const int tid = threadIdx.x;
  const int batch = blockIdx.x >> 7;
  const int p0 = (blockIdx.x & 127) * 128;
  for (int i = tid; i < 128 * 128; i += 256) {
    int c = i >> 7, px = i & 127;
    Xl[XI(c, px)] = dyn[((size_t)batch * D2 + c) * HWN + p0 + px];
  }
  for (int i = tid; i < 512 * 32; i += 256) Fl[(i >> 5) * 36 + (i & 31)] = fdcw[i];
  for (int i = tid; i < 512; i += 256) wtl[i] = wtw[i];
  __syncthreads();
  if (tid < 128) {
    int px = tid;
    float lg[GG];
    for (int g = 0; g < GG; ++g) {
      float s = wtb[g];
      for (int c = 0; c < 128; ++c) s += wtl[g * 128 + c] * Xl[XI(c, px)];
      lg[g] = s;
    }
    float mx = fmaxf(fmaxf(lg[0], lg[1]), fmaxf(lg[2], lg[3]));
    float den = 0.f;
    for (int g = 0; g < GG; ++g) { lg[g] = expf(lg[g] - mx); den += lg[g]; }
    for (int g = 0; g < GG; ++g) dyw[g * 128 + px] = lg[g] / den;
  }
  __syncthreads();
  const int wave = tid >> 5, lane = tid & 31;
  const int lh = lane & 15, kh = (lane >> 4) * 2, hi = lane >> 4;
  const int m0 = wave * 16;
  v8f zero = {};
  for (int n = 0; n < 8; ++n) {
    const int col = n * 16 + lh;
    v8f acc[GG];
#pragma unroll
    for (int g = 0; g < GG; ++g) acc[g] = zero;
    // software-pipelined across the whole flattened (g,k) range
    v2f a_c = ld2(&Fl[(m0 + lh) * 36 + kh]);
    v2f b_c = ld2(&Xl[XI(kh, col)]);
#pragma unroll
    for (int g = 0; g < GG; ++g)
#pragma unroll
      for (int klc = 0; klc < GK; klc += 4) {
        const int kn = g * GK + klc + 4; // next flattened K position
        v2f a_n = a_c, b_n = b_c;
        if (kn < 128) {
          const int gn = kn >> 5, kln = kn & 31;
          a_n = ld2(&Fl[((gn * 128) + m0 + lh) * 36 + kln + kh]);
          b_n = ld2(&Xl[XI(kn + kh, col)]);
        }
        acc[g] = wmma4(a_c, b_c, acc[g]);
        a_c = a_n; b_c = b_n;
      }
    float res[8];
#pragma unroll
    for (int r = 0; r < 8; ++r) res[r] = 0.f;
#pragma unroll
    for (int g = 0; g < GG; ++g) {
      float sc = dyw[g * 128 + col];
#pragma unroll
      for (int r = 0; r < 8; ++r) res[r] += acc[g][r] * sc;
    }
#pragma unroll
    for (int r = 0; r < 8; ++r) {
      int o = m0 + r + hi * 8;
      float bsum = 0.f;
      for (int g = 0; g < GG; ++g) bsum += dyw[g * 128 + col] * fdcb[g * 128 + o];
      size_t off = ((size_t)batch * D2 + o) * HWN + p0 + col;
      dct_fuse[off] = res[r] + bsum + dct_fuse[off];
    }
  }
}

// ---------------------------------------------------------------------------
// conv1x1 128->64 on (out + x0), exact GELU. grid = B*(HW/128), 256 thr.
__global__ void k_cf(const float* __restrict__ outb, const float* __restrict__ x0,
                     const float* __restrict__ w, const float* __restrict__ bias,
                     float* __restrict__ xf) {
  extern __shared__ float sm[];
  float* Wl = sm;            // [64][132] row-major
  float* Xl = sm + 64 * 132; // interleaved 64 pairs x 288
  const int tid = threadIdx.x;
  const int batch = blockIdx.x >> 7;
  const int p0 = (blockIdx.x & 127) * 128;
  for (int i = tid; i < 64 * 128; i += 256) Wl[(i >> 7) * 132 + (i & 127)] = w[i];
  for (int i = tid; i < 128 * 128; i += 256) {
    int c = i >> 7, px = i & 127;
    size_t off = ((size_t)batch * D2 + c) * HWN + p0 + px;
    Xl[XI(c, px)] = outb[off] + x0[off];
  }
  __syncthreads();
  const int wave = tid >> 5, lane = tid & 31;
  const int lh = lane & 15, kh = (lane >> 4) * 2, hi = lane >> 4;
  const int m0 = (wave & 3) * 16;
  const int nbase = (wave >> 2) * 4;
  v8f zero = {};
  v8f acc[4];
#pragma unroll
  for (int n = 0; n < 4; ++n) acc[n] = zero;
  v2f a_c = ld2(&Wl[(m0 + lh) * 132 + kh]);
  v2f bf_c[4];
#pragma unroll
  for (int n = 0; n < 4; ++n) bf_c[n] = ld2(&Xl[XI(kh, (nbase + n) * 16 + lh)]);
  for (int k0 = 0; k0 < 124; k0 += 4) {
    const int kn = k0 + 4;
    v2f a_n = ld2(&Wl[(m0 + lh) * 132 + kn + kh]);
    v2f bf_n[4];
#pragma unroll
    for (int n = 0; n < 4; ++n) bf_n[n] = ld2(&Xl[XI(kn + kh, (nbase + n) * 16 + lh)]);
#pragma unroll
    for (int n = 0; n < 4; ++n) acc[n] = wmma4(a_c, bf_c[n], acc[n]);
    a_c = a_n;
#pragma unroll
    for (int n = 0; n < 4; ++n) bf_c[n] = bf_n[n];
  }
#pragma unroll
  for (int n = 0; n < 4; ++n) acc[n] = wmma4(a_c, bf_c[n], acc[n]);
#pragma unroll
  for (int n = 0; n < 4; ++n)
#pragma unroll
    for (int r = 0; r < 8; ++r) {
      int o = m0 + r + hi * 8, px = (nbase + n) * 16 + lh;
      float v = acc[n][r] + bias[o];
      xf[((size_t)batch * DD + o) * HWN + p0 + px] = gelu_exact(v);
    }
}

// ---------------------------------------------------------------------------
// cpatten stage 1: v = silu(bn(v_w*xf + v_b)) via WMMA; klogit = silu(bn(k_w*xf)).
__global__ void k_vk(const float* __restrict__ xf, const float* __restrict__ vw,
                     const float* __restrict__ vb, const float* __restrict__ vg,
                     const float* __restrict__ vbb, const float* __restrict__ vrm,
                     const float* __restrict__ vrv, const float* __restrict__ kw,
                     const float* __restrict__ kb, const float* __restrict__ kg,
                     const float* __restrict__ kbb, const float* __restrict__ krm,
                     const float* __restrict__ krv, float* __restrict__ v,
                     float* __restrict__ kl) {
  extern __shared__ float sm[];
  float* Wl = sm;           // [64][68]
  float* Xl = sm + 64 * 68; // interleaved 32 pairs x 288
  const int tid = threadIdx.x;
  const int batch = blockIdx.x >> 7;
  const int p0 = (blockIdx.x & 127) * 128;
  for (int i = tid; i < 64 * 64; i += 256) Wl[(i >> 6) * 68 + (i & 63)] = vw[i];
  for (int i = tid; i < 64 * 128; i += 256) {
    int c = i >> 7, px = i & 127;
    Xl[XI(c, px)] = xf[((size_t)batch * DD + c) * HWN + p0 + px];
  }
  __syncthreads();
  if (tid < 128) {
    float s = kb[0];
    for (int c = 0; c < 64; ++c) s += kw[c] * Xl[XI(c, tid)];
    float scale = kg[0] * rsqrtf(krv[0] + EPS);
    s = (s - krm[0]) * scale + kbb[0];
    kl[(size_t)batch * HWN + p0 + tid] = silu(s);
  }
  const int wave = tid >> 5, lane = tid & 31;
  const int lh = lane & 15, kh = (lane >> 4) * 2, hi = lane >> 4;
  const int m0 = (wave & 3) * 16;
  const int nbase = (wave >> 2) * 4;
  v8f zero = {};
  v8f acc[4];
#pragma unroll
  for (int n = 0; n < 4; ++n) acc[n] = zero;
  v2f a_c = ld2(&Wl[(m0 + lh) * 68 + kh]);
  v2f bf_c[4];
#pragma unroll
  for (int n = 0; n < 4; ++n) bf_c[n] = ld2(&Xl[XI(kh, (nbase + n) * 16 + lh)]);
  for (int k0 = 0; k0 < 60; k0 += 4) {
    const int kn = k0 + 4;
    v2f a_n = ld2(&Wl[(m0 + lh) * 68 + kn + kh]);
    v2f bf_n[4];
#pragma unroll
    for (int n = 0; n < 4; ++n) bf_n[n] = ld2(&Xl[XI(kn + kh, (nbase + n) * 16 + lh)]);
#pragma unroll
    for (int n = 0; n < 4; ++n) acc[n] = wmma4(a_c, bf_c[n], acc[n]);
    a_c = a_n;
#pragma unroll
    for (int n = 0; n < 4; ++n) bf_c[n] = bf_n[n];
  }
#pragma unroll
  for (int n = 0; n < 4; ++n) acc[n] = wmma4(a_c, bf_c[n], acc[n]);
#pragma unroll
  for (int n = 0; n < 4; ++n)
#pragma unroll
    for (int r = 0; r < 8; ++r) {
      int o = m0 + r + hi * 8, px = (nbase + n) * 16 + lh;
      float val = acc[n][r] + vb[o];
      float scale = vg[o] * rsqrtf(vrv[o] + EPS);
      val = (val - vrm[o]) * scale + vbb[o];
      v[((size_t)batch * DD + o) * HWN + p0 + px] = silu(val);
    }
}

// per-batch softmax stats of klogit. grid = B.
__global__ void k_kred(const float* __restrict__ kl, float* __restrict__ kstat) {
  __shared__ float red[256];
  const int b = blockIdx.x, tid = threadIdx.x;
  const float* p = kl + (size_t)b * HWN;
  float mx = -1e30f;
  for (int i = tid; i < HWN; i += 256) mx = fmaxf(mx, p[i]);
  red[tid] = mx;
  __syncthreads();
  for (int s = 128; s > 0; s >>= 1) {
    if (tid < s) red[tid] = fmaxf(red[tid], red[tid + s]);
    __syncthreads();
  }
  mx = red[0];
  __syncthreads();
  float sum = 0.f;
  for (int i = tid; i < HWN; i += 256) sum += expf(p[i] - mx);
  red[tid] = sum;
  __syncthreads();
  for (int s = 128; s > 0; s >>= 1) {
    if (tid < s) red[tid] += red[tid + s];
    __syncthreads();
  }
  if (tid == 0) { kstat[b * 2] = mx; kstat[b * 2 + 1] = red[0]; }
}

// per-(b,c) reductions: sum, sumsq, max of xf; kv = sum_hw v*softmax(kl). grid=B*64.
__global__ void k_cred(const float* __restrict__ xf, const float* __restrict__ v,
                       const float* __restrict__ kl, const float* __restrict__ kstat,
                       float* __restrict__ cstat) {
  __shared__ float red[256];
  const int bc = blockIdx.x, tid = threadIdx.x;
  const int b = bc >> 6;
  const float* px = xf + (size_t)bc * HWN;
  const float* pv = v + (size_t)bc * HWN;
  const float* pk = kl + (size_t)b * HWN;
  const float kmax = kstat[b * 2], kinv = 1.f / kstat[b * 2 + 1];
  float s1 = 0.f, s2 = 0.f, mx = -1e30f, kv = 0.f;
  for (int i = tid; i < HWN; i += 256) {
    float x = px[i];
    s1 += x; s2 += x * x; mx = fmaxf(mx, x);
    kv += pv[i] * expf(pk[i] - kmax);
  }
  red[tid] = s1; __syncthreads();
  for (int s = 128; s > 0; s >>= 1) { if (tid < s) red[tid] += red[tid + s]; __syncthreads(); }
  if (tid == 0) cstat[bc * 4 + 0] = red[0];
  __syncthreads();
  red[tid] = s2; __syncthreads();
  for (int s = 128; s > 0; s >>= 1) { if (tid < s) red[tid] += red[tid + s]; __syncthreads(); }
  if (tid == 0) cstat[bc * 4 + 1] = red[0];
  __syncthreads();
  red[tid] = mx; __syncthreads();
  for (int s = 128; s > 0; s >>= 1) { if (tid < s) red[tid] = fmaxf(red[tid], red[tid + s]); __syncthreads(); }
  if (tid == 0) cstat[bc * 4 + 2] = red[0];
  __syncthreads();
  red[tid] = kv; __syncthreads();
  for (int s = 128; s > 0; s >>= 1) { if (tid < s) red[tid] += red[tid + s]; __syncthreads(); }
  if (tid == 0) cstat[bc * 4 + 3] = red[0] * kinv;
}

// per-batch vectors: avg=softmax(mean), mxs=softmax(max), std*gamma, mkv=m_w*kv+m_b.
__global__ void k_vec(const float* __restrict__ cstat, const float* __restrict__ gamma,
                      const float* __restrict__ mw, const float* __restrict__ mb,
                      float* __restrict__ vec) {
  __shared__ float sh[64];
  __shared__ float kvsh[64];
  const int b = blockIdx.x, c = threadIdx.x;
  const int i4 = (b * 64 + c) * 4;
  float s1 = cstat[i4], s2 = cstat[i4 + 1], mx = cstat[i4 + 2], kv = cstat[i4 + 3];
  float mean = s1 / (float)HWN;
  float var = (s2 - (float)HWN * mean * mean) / ((float)HWN - 1.f);
  float stdv = sqrtf(fmaxf(var, 0.f)) * gamma[c];
  sh[c] = mean; __syncthreads();
  for (int s = 32; s > 0; s >>= 1) { if (c < s) sh[c] = fmaxf(sh[c], sh[c + s]); __syncthreads(); }
  float m1 = sh[0]; __syncthreads();
  float e1 = expf(mean - m1);
  sh[c] = e1; __syncthreads();
  for (int s = 32; s > 0; s >>= 1) { if (c < s) sh[c] += sh[c + s]; __syncthreads(); }
  float avg = e1 / sh[0]; __syncthreads();
  sh[c] = mx; __syncthreads();
  for (int s = 32; s > 0; s >>= 1) { if (c < s) sh[c] = fmaxf(sh[c], sh[c + s]); __syncthreads(); }
  float m2 = sh[0]; __syncthreads();
  float e2 = expf(mx - m2);
  sh[c] = e2; __syncthreads();
  for (int s = 32; s > 0; s >>= 1) { if (c < s) sh[c] += sh[c + s]; __syncthreads(); }
  float mxs = e2 / sh[0];
  kvsh[c] = kv; __syncthreads();
  float mkv = mb[c];
  for (int j = 0; j < 64; ++j) mkv += mw[c * 64 + j] * kvsh[j];
  vec[i4] = avg; vec[i4 + 1] = mxs; vec[i4 + 2] = stdv; vec[i4 + 3] = mkv;
}

// final: q dots, m3 path, out = xf + mkv*sigmoid(m3). grid = B*HW/256.
__global__ void k_final(const float* __restrict__ xf, const float* __restrict__ v,
                        const float* __restrict__ vec, const float* __restrict__ m3w,
                        const float* __restrict__ m3b, const float* __restrict__ m3g,
                        const float* __restrict__ m3bb, const float* __restrict__ m3rm,
                        const float* __restrict__ m3rv, float* __restrict__ out) {
  __shared__ float avg[64], mxs[64], stdv[64], mkv[64];
  const int b = blockIdx.x >> 6;
  const int p0 = (blockIdx.x & 63) * 256;
  const int tid = threadIdx.x;
  if (tid < 64) {
    avg[tid] = vec[(b * 64 + tid) * 4 + 0];
    mxs[tid] = vec[(b * 64 + tid) * 4 + 1];
    stdv[tid] = vec[(b * 64 + tid) * 4 + 2];
    mkv[tid] = vec[(b * 64 + tid) * 4 + 3];
  }
  __syncthreads();
  const int pxg = p0 + tid;
  float qa = 0.f, qm = 0.f, qs = 0.f;
  for (int c = 0; c < 64; ++c) {
    float vv = v[((size_t)b * DD + c) * HWN + pxg];
    qa += avg[c] * vv; qm += mxs[c] * vv; qs += stdv[c] * vv;
  }
  float m3 = m3w[0] * qa + m3w[1] * qm + m3w[2] * qs + m3b[0];
  float scale = m3g[0] * rsqrtf(m3rv[0] + EPS);
  m3 = (m3 - m3rm[0]) * scale + m3bb[0];
  m3 = silu(m3);
  float sg = 1.f / (1.f + expf(-m3));
  for (int c = 0; c < 64; ++c) {
    size_t off = ((size_t)b * DD + c) * HWN + pxg;
    out[off] = xf[off] + mkv[c] * sg;
  }
}

// ---------------------------------------------------------------------------
extern "C" void kernel_launch(void* const* d_in, const int* in_sizes, int n_in,
                              void* d_out, int out_size, void* d_ws, size_t ws_size,
                              hipStream_t stream) {
  (void)in_sizes; (void)n_in; (void)out_size; (void)ws_size;
  const float* x      = (const float*)d_in[0];
  const float* ci_w   = (const float*)d_in[1];
  const float* ci_b   = (const float*)d_in[2];
  const float* cf_w   = (const float*)d_in[3];
  const float* cf_b   = (const float*)d_in[4];
  const float* beta   = (const float*)d_in[5];
  const float* wt_w   = (const float*)d_in[6];
  const float* wt_b   = (const float*)d_in[7];
  const float* fdc_w  = (const float*)d_in[8];
  const float* fdc_b  = (const float*)d_in[9];
  const float* fpe_w  = (const float*)d_in[10];
  const float* fpe_b  = (const float*)d_in[11];
  const float* kw     = (const float*)d_in[12];
  const float* kb     = (const float*)d_in[13];
  const float* k_bn_g = (const float*)d_in[14];
  const float* k_bn_b = (const float*)d_in[15];
  const float* k_bn_rm= (const float*)d_in[16];
  const float* k_bn_rv= (const float*)d_in[17];
  const float* v_w    = (const float*)d_in[18];
  const float* v_b    = (const float*)d_in[19];
  const float* v_bn_g = (const float*)d_in[20];
  const float* v_bn_b = (const float*)d_in[21];
  const float* v_bn_rm= (const float*)d_in[22];
  const float* v_bn_rv= (const float*)d_in[23];
  const float* m_w    = (const float*)d_in[24];
  const float* m_b    = (const float*)d_in[25];
  const float* m3_w   = (const float*)d_in[26];
  const float* m3_b   = (const float*)d_in[27];
  const float* m3_bn_g = (const float*)d_in[28];
  const float* m3_bn_b = (const float*)d_in[29];
  const float* m3_bn_rm= (const float*)d_in[30];
  const float* m3_bn_rv= (const float*)d_in[31];
  const float* gamma  = (const float*)d_in[32];

  const size_t NBIG = (size_t)BB * D2 * HWN;  // 16,777,216
  const size_t NHALF = (size_t)BB * DD * HWN; // 8,388,608
  float* ws   = (float*)d_ws;
  float* x1   = ws;            // x1 / x0 (kept until cf conv)
  float* dctb = x1 + NBIG;     // dct_feat, then fuse (in place)
  float* dynb = dctb + NBIG;   // dct_dyn, then inverse-DCT output (reuse)
  float* xfb  = dynb + NBIG;
  float* vbuf = xfb + NHALF;
  float* klb  = vbuf + NHALF;
  float* dmat = klb + (size_t)BB * HWN;
  float* filt = dmat + 16384;
  float* kst  = filt + 16384;
  float* cst  = kst + 16;
  float* vecb = cst + BB * 64 * 4;

  k_init_dct<<<64, 256, 0, stream>>>(dmat);
  k_init_filter<<<1, 256, 0, stream>>>(beta, filt);
  k_ci<<<BB * (HWN / 128), 256, (128 * 68 + 32 * 288) * 4, stream>>>(x, ci_w, ci_b, x1);
  k_dct<<<BB * D2, 256, 2 * 18432 * 4, stream>>>(x1, dctb, dmat, filt, 0);
  k_fpe<<<(int)(NBIG / 256), 256, 0, stream>>>(dctb, fpe_w, fpe_b, dynb);
  k_mix<<<BB * (HWN / 128), 256, (18432 + 512 * 36 + 1024) * 4, stream>>>(
      dynb, dctb, wt_w, wt_b, fdc_w, fdc_b);
  k_dct<<<BB * D2, 256, 2 * 18432 * 4, stream>>>(dctb, dynb, dmat, filt, 1);
  k_cf<<<BB * (HWN / 128), 256, (64 * 132 + 64 * 288) * 4, stream>>>(dynb, x1, cf_w,
                                                                     cf_b, xfb);
  k_vk<<<BB * (HWN / 128), 256, (64 * 68 + 32 * 288) * 4, stream>>>(
      xfb, v_w, v_b, v_bn_g, v_bn_b, v_bn_rm, v_bn_rv, kw, kb, k_bn_g, k_bn_b,
      k_bn_rm, k_bn_rv, vbuf, klb);
  k_kred<<<BB, 256, 0, stream>>>(klb, kst);
  k_cred<<<BB * 64, 256, 0, stream>>>(xfb, vbuf, klb, kst, cst);
  k_vec<<<BB, 64, 0, stream>>>(cst, gamma, m_w, m_b, vecb);
  k_final<<<BB * (HWN / 256), 256, 0, stream>>>(xfb, vbuf, vecb, m3_w, m3_b, m3_bn_g,
                                                m3_bn_b, m3_bn_rm, m3_bn_rv,
                                                (float*)d_out);
}